// GATNet_24154896072822
// MI455X (gfx1250) — compile-verified
//
#include <hip/hip_runtime.h>
#include <hip/hip_bf16.h>

#define NNODES 50000
#define NEDGES 500000
#define IN_C   128
#define HID_C  64
#define OUT_C  128
#define HEADS  4

typedef float v2f __attribute__((ext_vector_type(2)));
typedef float v8f __attribute__((ext_vector_type(8)));

// ---------------------------------------------------------------------------
// GEMM: Out[N, NCOL] = act(A[N, K]) @ W[K, NCOL] + bias, via V_WMMA_F32_16X16X4_F32.
// One wave32 per 16x16 output tile. A-fragment layout (ISA 7.12.2, 32-bit 16x4):
//   lanes 0-15: M = lane;  VGPR0 = A[M][k+0], VGPR1 = A[M][k+1]
//   lanes16-31: M = lane-16; VGPR0 = A[M][k+2], VGPR1 = A[M][k+3]
// B-fragment (4x16) mirrored: VGPR0 = W[k+2*hi][col], VGPR1 = W[k+2*hi+1][col].
// C/D: VGPR i holds row (i + 8*hi), col = lane&15.
// ---------------------------------------------------------------------------
template<int K, int NCOL, bool RELU_IN>
__global__ __launch_bounds__(256)
void gat_gemm_wmma(const float* __restrict__ A, const float* __restrict__ W,
                   const float* __restrict__ bias, float* __restrict__ Out) {
  const int lane    = threadIdx.x & 31;
  const int wave    = threadIdx.x >> 5;
  const int rowTile = blockIdx.x;
  const int colTile = blockIdx.y * 8 + wave;
  const int hi = lane >> 4;        // 0: K pair {0,1}; 1: K pair {2,3}
  const int lo = lane & 15;
  const int row = rowTile * 16 + lo;
  const int col = colTile * 16 + lo;
  const float* __restrict__ arow = A + (size_t)row * K;

  v8f c = {0.f, 0.f, 0.f, 0.f, 0.f, 0.f, 0.f, 0.f};
#pragma unroll 4
  for (int k = 0; k < K; k += 4) {
    const int ka = k + 2 * hi;
    float2 av = *(const float2*)(arow + ka);
    if (RELU_IN) { av.x = fmaxf(av.x, 0.f); av.y = fmaxf(av.y, 0.f); }
    v2f a = {av.x, av.y};
    v2f b = { W[(size_t)ka * NCOL + col],
              W[(size_t)(ka + 1) * NCOL + col] };
    c = __builtin_amdgcn_wmma_f32_16x16x4_f32(
        /*neg_a=*/false, a, /*neg_b=*/false, b,
        /*c_mod=*/(short)0, c, /*reuse_a=*/false, /*reuse_b=*/false);
  }

  const float bv = bias[col];
  float* __restrict__ orow = Out + (size_t)(rowTile * 16 + hi * 8) * NCOL + col;
#pragma unroll
  for (int i = 0; i < 8; ++i)
    orow[(size_t)i * NCOL] = c[i] + bv;
}

// ---------------------------------------------------------------------------
// Edge kernel: one wave32 per edge. Lane = (head g = lane>>3, chunk k = lane&7).
// Each lane gathers CH = D/8 floats of h[src][g][k*CH ..], computes partial
// attention dot, 8-lane shfl reduction -> per-head score; leaky_relu, *w,
// softmax over 4 heads via shfl_xor(8/16); scales message, reduces over heads
// (folding the final mean-over-heads /4), and lanes g==0 scatter-add D floats
// to out[dst] with global_atomic_add_f32 (no return).
// ---------------------------------------------------------------------------
template<int D>
__global__ __launch_bounds__(256)
void gat_edge(const float* __restrict__ h, const float* __restrict__ att,
              const int* __restrict__ src, const int* __restrict__ dst,
              const float* __restrict__ ew, float* __restrict__ out) {
  constexpr int CH  = D / 8;   // floats per lane
  constexpr int CH4 = CH / 4;  // float4 per lane
  const int lane = threadIdx.x & 31;
  const int e    = blockIdx.x * 8 + (threadIdx.x >> 5);
  if (e >= NEDGES) return;
  const int g = lane >> 3;
  const int k = lane & 7;
  const int s = src[e];
  const int t = dst[e];

  const float4* __restrict__ xp =
      (const float4*)(h + (size_t)s * (HEADS * D) + g * D + k * CH);
  const float4* __restrict__ ap = (const float4*)(att + g * D + k * CH);

  float4 v[CH4];
  float part = 0.f;
#pragma unroll
  for (int i = 0; i < CH4; ++i) {
    v[i] = xp[i];
    float4 a = ap[i];
    part += v[i].x * a.x + v[i].y * a.y + v[i].z * a.z + v[i].w * a.w;
  }
  // reduce dot across the 8 lanes of this head
  part += __shfl_xor(part, 1, 32);
  part += __shfl_xor(part, 2, 32);
  part += __shfl_xor(part, 4, 32);

  float sc = (part > 0.f ? part : 0.2f * part) * ew[e];   // leaky_relu * w
  // softmax over the 4 heads (values uniform within each 8-lane group)
  float m = fmaxf(sc, __shfl_xor(sc, 8, 32));
  m       = fmaxf(m, __shfl_xor(m, 16, 32));
  float ex  = __expf(sc - m);
  float den = ex + __shfl_xor(ex, 8, 32);
  den      += __shfl_xor(den, 16, 32);
  const float alpha = (ex / den) * 0.25f;  // fold mean over heads (1/H)

  float* __restrict__ op = out + (size_t)t * D + k * CH;
#pragma unroll
  for (int i = 0; i < CH4; ++i) {
    float4 mv;
    mv.x = v[i].x * alpha; mv.y = v[i].y * alpha;
    mv.z = v[i].z * alpha; mv.w = v[i].w * alpha;
    // sum over the 4 heads: shfl_xor across group bits
    mv.x += __shfl_xor(mv.x, 8, 32);  mv.x += __shfl_xor(mv.x, 16, 32);
    mv.y += __shfl_xor(mv.y, 8, 32);  mv.y += __shfl_xor(mv.y, 16, 32);
    mv.z += __shfl_xor(mv.z, 8, 32);  mv.z += __shfl_xor(mv.z, 16, 32);
    mv.w += __shfl_xor(mv.w, 8, 32);  mv.w += __shfl_xor(mv.w, 16, 32);
    if (g == 0) {
      unsafeAtomicAdd(op + i * 4 + 0, mv.x);
      unsafeAtomicAdd(op + i * 4 + 1, mv.y);
      unsafeAtomicAdd(op + i * 4 + 2, mv.z);
      unsafeAtomicAdd(op + i * 4 + 3, mv.w);
    }
  }
}

extern "C" void kernel_launch(void* const* d_in, const int* in_sizes, int n_in,
                              void* d_out, int out_size, void* d_ws, size_t ws_size,
                              hipStream_t stream) {
  const float* x    = (const float*)d_in[0];
  const int*   ei   = (const int*)d_in[1];
  const float* ewt  = (const float*)d_in[2];
  const float* W1   = (const float*)d_in[3];
  const float* b1   = (const float*)d_in[4];
  const float* att1 = (const float*)d_in[5];
  const float* W2   = (const float*)d_in[6];
  const float* b2   = (const float*)d_in[7];
  const float* att2 = (const float*)d_in[8];
  const float* W3   = (const float*)d_in[9];
  const float* b3   = (const float*)d_in[10];
  const float* att3 = (const float*)d_in[11];
  float* out = (float*)d_out;

  const int* src = ei;            // edge_index[0]
  const int* dst = ei + NEDGES;   // edge_index[1]

  float* hbuf   = (float*)d_ws;                          // N x 512 (reused)
  float* aggbuf = hbuf + (size_t)NNODES * (HEADS * OUT_C); // N x 64 (reused)

  const dim3 blk(256);
  const dim3 gemm_grid12(NNODES / 16, (HEADS * HID_C) / (16 * 8)); // 3125 x 2
  const dim3 gemm_grid3 (NNODES / 16, (HEADS * OUT_C) / (16 * 8)); // 3125 x 4
  const dim3 edge_grid(NEDGES / 8);                                 // 62500

  // ---- Layer 1: IN -> H*HID ----
  gat_gemm_wmma<IN_C, HEADS * HID_C, false>
      <<<gemm_grid12, blk, 0, stream>>>(x, W1, b1, hbuf);
  hipMemsetAsync(aggbuf, 0, (size_t)NNODES * HID_C * sizeof(float), stream);
  gat_edge<HID_C><<<edge_grid, blk, 0, stream>>>(hbuf, att1, src, dst, ewt, aggbuf);

  // ---- Layer 2: HID -> H*HID (ReLU folded into A load) ----
  gat_gemm_wmma<HID_C, HEADS * HID_C, true>
      <<<gemm_grid12, blk, 0, stream>>>(aggbuf, W2, b2, hbuf);
  hipMemsetAsync(aggbuf, 0, (size_t)NNODES * HID_C * sizeof(float), stream);
  gat_edge<HID_C><<<edge_grid, blk, 0, stream>>>(hbuf, att2, src, dst, ewt, aggbuf);

  // ---- Layer 3: HID -> H*OUT ----
  gat_gemm_wmma<HID_C, HEADS * OUT_C, true>
      <<<gemm_grid3, blk, 0, stream>>>(aggbuf, W3, b3, hbuf);
  hipMemsetAsync(out, 0, (size_t)NNODES * OUT_C * sizeof(float), stream);
  gat_edge<OUT_C><<<edge_grid, blk, 0, stream>>>(hbuf, att3, src, dst, ewt, out);
}